// SelfAttentionHead_18949395710575
// MI455X (gfx1250) — compile-verified
//
#include <hip/hip_runtime.h>
#include <hip/hip_bf16.h>

// ---------------------------------------------------------------------------
// Fused causal self-attention head, fp32, CDNA5 (gfx1250, wave32).
//   B=16, T=1024, C=768, H=64.
// Kernel 1: K/Q/V = x @ W{k,q,v}  via V_WMMA_F32_16X16X4_F32, with the x tile
//           staged into LDS by the Tensor Data Mover (tensor_load_to_lds)
//           using D# pad_enable to create a bank-conflict-free padded layout.
// Kernel 2: flash-style causal softmax(Q K^T / 8) V, one wave per 16-query
//           tile, online softmax, P-tile transposed C/D->A layout via LDS.
// ---------------------------------------------------------------------------

typedef float v2f __attribute__((ext_vector_type(2)));
typedef float v8f __attribute__((ext_vector_type(8)));
typedef unsigned int v4u __attribute__((ext_vector_type(4)));
typedef int v4i __attribute__((ext_vector_type(4)));
typedef int v8i __attribute__((ext_vector_type(8)));

#define B_DIM 16
#define T_DIM 1024
#define C_DIM 768
#define H_DIM 64
#define BTH   (B_DIM * T_DIM * H_DIM)   // 1,048,576 floats per matrix

// D(16x16,f32) = A(16x4,f32) x B(4x16,f32) + C
static __device__ __forceinline__ v8f wmma_f32x4(v2f a, v2f b, v8f c) {
  // args: (neg_a, A, neg_b, B, c_mod, C, reuse_a, reuse_b)
  return __builtin_amdgcn_wmma_f32_16x16x4_f32(false, a, false, b,
                                               (short)0, c, false, false);
}

static __device__ __forceinline__ float rowmax16(float v) {
  v = fmaxf(v, __shfl_xor(v, 1, 32));
  v = fmaxf(v, __shfl_xor(v, 2, 32));
  v = fmaxf(v, __shfl_xor(v, 4, 32));
  v = fmaxf(v, __shfl_xor(v, 8, 32));
  return v;
}
static __device__ __forceinline__ float rowsum16(float v) {
  v += __shfl_xor(v, 1, 32);
  v += __shfl_xor(v, 2, 32);
  v += __shfl_xor(v, 4, 32);
  v += __shfl_xor(v, 8, 32);
  return v;
}

// ---------------------------------------------------------------------------
// Kernel 1: projections.  grid.x = (B*T)/16 = 1024 row tiles, 384 threads
// (12 waves).  Wave w: matrix = w/4 (0:K 1:Q 2:V), col-tile = w%4.
// The 16x768 x tile (48 KB, contiguous) is DMA'd into LDS by the TDM with
// 4-DWORD padding every 256 DWORDs -> effective row stride 780 floats
// (780 mod 64 = 12, coprime stride over 16 rows => conflict-free A reads).
// ---------------------------------------------------------------------------
__global__ __launch_bounds__(384) void proj_kqv_kernel(
    const float* __restrict__ x,
    const float* __restrict__ Wk, const float* __restrict__ Wq,
    const float* __restrict__ Wv,
    float* __restrict__ Ko, float* __restrict__ Qo, float* __restrict__ Vo) {
  constexpr int LS = 780;                    // padded row stride (floats)
  __shared__ float xs[12480];                // 12288 data + 192 pad DWORDs

  const int tid = threadIdx.x;
  const int rowbase = blockIdx.x * 16;

  // ---- TDM: one wave issues tensor_load_to_lds for the whole tile ----
  if (tid < 32) {
    const unsigned long long ga =
        (unsigned long long)(uintptr_t)(x + (size_t)rowbase * C_DIM);
    const unsigned int lds_off = (unsigned int)(uintptr_t)xs; // addr[31:0]

    // D# group 0: count=1 | lds_addr | global_addr[56:0] | type=2
    v4u g0;
    g0.x = 1u;                                           // count=1, user mode
    g0.y = lds_off;                                      // lds_addr
    g0.z = (unsigned int)(ga & 0xFFFFFFFFu);             // global_addr[31:0]
    g0.w = (unsigned int)((ga >> 32) & 0x01FFFFFFu)      // global_addr[56:32]
           | (2u << 30);                                 // type = 2 ("image")

    // D# group 1: data_size=4B(2) | pad_enable | pad_interval=7(256dw)
    //             | pad_amount=3(4dw); dims/strides for 12288x1 tile.
    v8i g1;
    g1[0] = (int)((2u << 16) | (1u << 20) | (7u << 22) | (3u << 25));
    g1[1] = (int)(12288u << 16);   // tensor_dim0[15:0] = 12288
    g1[2] = (int)(1u << 16);       // tensor_dim1[15:0] = 1
    g1[3] = (int)(12288u << 16);   // tile_dim0 = 12288
    g1[4] = 1;                     // tile_dim1 = 1 (tile_dim2 = 0)
    g1[5] = 12288;                 // tensor_dim0_stride[31:0]
    g1[6] = (int)(12288u << 16);   // tensor_dim1_stride[15:0]
    g1[7] = 0;

    v4i g2 = {};                   // dims 2/3 unused
    v4i g3 = {};
    v8i g4 = {};                   // extra operand of the 6-arg builtin

    __builtin_amdgcn_tensor_load_to_lds(g0, g1, g2, g3, g4, 0);
    __builtin_amdgcn_s_wait_tensorcnt(0);
  }
  __syncthreads();

  const int wid  = tid >> 5;
  const int lane = tid & 31;
  const int half = lane >> 4;                // K sub-select for A/B frags
  const int ln   = lane & 15;                // M (A) / N (B) index
  const int mat  = wid >> 2;                 // 0:K  1:Q  2:V
  const int ct   = wid & 3;                  // column tile 0..3

  const float* W   = (mat == 0) ? Wk : (mat == 1 ? Wq : Wv);
  float*       Out = (mat == 0) ? Ko : (mat == 1 ? Qo : Vo);
  const int col = ct * 16 + ln;

  v8f acc = {};
#pragma unroll 4
  for (int k0 = 0; k0 < C_DIM; k0 += 4) {
    const int kk = k0 + 2 * half;
    // Padded LDS index: 4 extra DWORDs per 256 stored (kk never crosses a
    // pad boundary relative to k0 since k0 % 4 == 0).
    const int aofs = ln * LS + kk + 4 * (k0 >> 8);
    // A frag: lane = row M, (half, .x/.y) = K   (ISA 16x4 f32 A layout)
    v2f a;
    a.x = xs[aofs];
    a.y = xs[aofs + 1];
    // B frag: lane = col N, (half, .x/.y) = K   (B = W, row-major C x H)
    v2f b;
    b.x = W[kk * H_DIM + col];
    b.y = W[(kk + 1) * H_DIM + col];
    acc = wmma_f32x4(a, b, acc);
  }

  // C/D layout: VGPR r, lane half -> row r + 8*half, lane&15 -> col.
#pragma unroll
  for (int r = 0; r < 8; ++r)
    Out[(size_t)(rowbase + r + 8 * half) * H_DIM + col] = acc[r];
}

// ---------------------------------------------------------------------------
// Kernel 2: causal attention.  4 waves / block, each wave owns one 16-query
// tile and iterates over key tiles 0..qt with online softmax.
// ---------------------------------------------------------------------------
__global__ __launch_bounds__(128) void attn_kernel(
    const float* __restrict__ Q, const float* __restrict__ K,
    const float* __restrict__ V, float* __restrict__ out) {
  constexpr int PS = 17;                     // padded P-tile row stride
  __shared__ float ps[4 * 16 * PS];

  const int tid  = threadIdx.x;
  const int wid  = tid >> 5;
  const int lane = tid & 31;
  const int half = lane >> 4;
  const int ln   = lane & 15;

  const int gw    = blockIdx.x * 4 + wid;    // global wave id, 0..1023
  const int batch = gw >> 6;
  const int qt    = gw & 63;                 // query tile within batch
  const int qbase = qt * 16;

  const size_t boff = (size_t)batch * T_DIM * H_DIM;
  const float* Qb = Q + boff;
  const float* Kb = K + boff;
  const float* Vb = V + boff;
  float*       Ob = out + boff;

  // Pre-load Q A-fragments, pre-scaled by 1/sqrt(H)=0.125.
  v2f qa[16];
#pragma unroll
  for (int s = 0; s < 16; ++s) {
    const int kk = 4 * s + 2 * half;
    const float* p = Qb + (size_t)(qbase + ln) * H_DIM + kk;
    qa[s].x = p[0] * 0.125f;
    qa[s].y = p[1] * 0.125f;
  }

  v8f o[4];
  o[0] = (v8f){};  o[1] = (v8f){};  o[2] = (v8f){};  o[3] = (v8f){};
  float mrow[8], lrow[8];
#pragma unroll
  for (int r = 0; r < 8; ++r) { mrow[r] = -__builtin_inff(); lrow[r] = 0.0f; }

  float* pws = ps + wid * 16 * PS;           // private P staging per wave

  for (int j = 0; j <= qt; ++j) {
    const float* Kt = Kb + (size_t)j * 16 * H_DIM;
    const float* Vt = Vb + (size_t)j * 16 * H_DIM;
    if (j < qt)  // pull next K tile toward the caches (global_prefetch_b8)
      __builtin_prefetch(Kb + (size_t)(j + 1) * 16 * H_DIM, 0, 1);

    // ---- S = (Q*0.125) @ K^T : 16 chained f32 WMMAs over H=64 ----
    v8f s8 = {};
#pragma unroll
    for (int st = 0; st < 16; ++st) {
      const int kk = 4 * st + 2 * half;
      const float* kp = Kt + (size_t)ln * H_DIM + kk;  // B[k][n] = K[n][k]
      v2f bK; bK.x = kp[0]; bK.y = kp[1];
      s8 = wmma_f32x4(qa[st], bK, s8);
    }

    // ---- causal mask on the diagonal tile ----
    if (j == qt) {
#pragma unroll
      for (int r = 0; r < 8; ++r) {
        const int M = r + 8 * half;          // query row within tile
        if (ln > M) s8[r] = -1.0e30f;        // key col ln > query row M
      }
    }

    // ---- online softmax (row r+8*half lives in element r of each lane) ----
    float sf[8], pv[8];
#pragma unroll
    for (int r = 0; r < 8; ++r) {
      const float rmax = rowmax16(s8[r]);
      const float mn   = fmaxf(mrow[r], rmax);
      const float fac  = __expf(mrow[r] - mn);
      const float p    = __expf(s8[r] - mn);
      lrow[r] = lrow[r] * fac + rowsum16(p);
      mrow[r] = mn;
      sf[r] = fac;
      pv[r] = p;
    }
#pragma unroll
    for (int r = 0; r < 8; ++r) {
      o[0][r] *= sf[r]; o[1][r] *= sf[r]; o[2][r] *= sf[r]; o[3][r] *= sf[r];
    }

    // ---- stage P (C/D layout) into LDS, re-read in A layout ----
#pragma unroll
    for (int r = 0; r < 8; ++r)
      pws[(r + 8 * half) * PS + ln] = pv[r];
    asm volatile("s_wait_dscnt 0" ::: "memory");  // in-wave store->load order

    // ---- O += P @ V : K=16 in 4 chunks x 4 column tiles ----
#pragma unroll
    for (int ch = 0; ch < 4; ++ch) {
      const int kk = 4 * ch + 2 * half;
      v2f a;
      a.x = pws[ln * PS + kk];
      a.y = pws[ln * PS + kk + 1];
#pragma unroll
      for (int t = 0; t < 4; ++t) {
        const int colv = t * 16 + ln;
        const float* vp = Vt + (size_t)kk * H_DIM + colv;
        v2f bV; bV.x = vp[0]; bV.y = vp[H_DIM];
        o[t] = wmma_f32x4(a, bV, o[t]);
      }
    }
    asm volatile("" ::: "memory");  // keep next iter's DS stores after reads
  }

  // ---- epilogue: out = O / l ----
  float inv[8];
#pragma unroll
  for (int r = 0; r < 8; ++r) inv[r] = 1.0f / lrow[r];
#pragma unroll
  for (int t = 0; t < 4; ++t)
#pragma unroll
    for (int r = 0; r < 8; ++r)
      Ob[(size_t)(qbase + r + 8 * half) * H_DIM + t * 16 + ln] =
          o[t][r] * inv[r];
}

// ---------------------------------------------------------------------------
extern "C" void kernel_launch(void* const* d_in, const int* in_sizes, int n_in,
                              void* d_out, int out_size, void* d_ws,
                              size_t ws_size, hipStream_t stream) {
  const float* x  = (const float*)d_in[0];
  const float* Wk = (const float*)d_in[1];
  const float* Wq = (const float*)d_in[2];
  const float* Wv = (const float*)d_in[3];

  float* wsf = (float*)d_ws;
  float* Kw = wsf;               // [B,T,H]
  float* Qw = wsf + BTH;         // [B,T,H]
  float* Vw = wsf + 2 * (size_t)BTH;

  // 1) projections: 1024 row tiles, 12 waves each
  proj_kqv_kernel<<<dim3((B_DIM * T_DIM) / 16), dim3(384), 0, stream>>>(
      x, Wk, Wq, Wv, Kw, Qw, Vw);

  // 2) attention: 1024 query-tile waves, 4 waves per block
  attn_kernel<<<dim3((B_DIM * (T_DIM / 16)) / 4), dim3(128), 0, stream>>>(
      Qw, Kw, Vw, (float*)d_out);
}